// EdgeSimilarityPooling_40286793236958
// MI455X (gfx1250) — compile-verified
//
#include <hip/hip_runtime.h>
#include <hip/hip_bf16.h>

// ---------------- problem constants ----------------
#define NPTS   16384
#define DIMF   64
#define NEDGE  262144
#define KNN    3
#define M_TOT  (NPTS * KNN + NEDGE)   // 311296 directed candidate edges
#define TWO_M  (2 * M_TOT)            // 622592 keys (fwd + rev)
#define OUT_ELEMS (2 * TWO_M)         // output is [2, 2M] int32
#define INVALID_KEY 0xFFFFFFFFu
#define COS_EPS 1e-8f
#define THRESH  0.1f
#define FBIG    3.4e38f

// ---------------- WMMA vector types ----------------
typedef __attribute__((ext_vector_type(2))) float v2f;  // A/B frag of 16x16x4 f32 (2 VGPRs/lane)
typedef __attribute__((ext_vector_type(8))) float v8f;  // C/D frag (8 VGPRs/lane)

// ---------------- branchless top-3 (ascending: s0 <= s1 <= s2) ----------------
__device__ __forceinline__ void top3_update(float s, int j,
                                            float& s0, float& s1, float& s2,
                                            int& i0, int& i1, int& i2) {
  // conditional replace of the worst slot, then bubble down: cmp+cndmask only
  bool c2 = s < s2;
  s2 = c2 ? s : s2;  i2 = c2 ? j : i2;
  bool c1 = s2 < s1;
  float ts = s1; int ti = i1;
  s1 = c1 ? s2 : s1;  i1 = c1 ? i2 : i1;
  s2 = c1 ? ts : s2;  i2 = c1 ? ti : i2;
  bool c0 = s1 < s0;
  ts = s0; ti = i0;
  s0 = c0 ? s1 : s0;  i0 = c0 ? i1 : i0;
  s1 = c0 ? ts : s1;  i1 = c0 ? ti : i1;
}

// ---------------- kernel 0: init output / counters ----------------
__global__ __launch_bounds__(256) void init_kernel(int* __restrict__ out,
                                                   unsigned* __restrict__ counts,
                                                   unsigned* __restrict__ cursors) {
  int t = blockIdx.x * blockDim.x + threadIdx.x;
  if (t < OUT_ELEMS) out[t] = -1;
  if (t < NPTS) { counts[t] = 0u; cursors[t] = 0u; }
}

// ---------------- kernel 1: row L2 norms of x ----------------
__global__ __launch_bounds__(256) void norm_kernel(const float* __restrict__ x,
                                                   float* __restrict__ nrm) {
  int i = blockIdx.x * blockDim.x + threadIdx.x;
  if (i >= NPTS) return;
  const float4* row = (const float4*)(x + (size_t)i * DIMF);
  float acc = 0.f;
#pragma unroll
  for (int q = 0; q < DIMF / 4; ++q) {
    float4 v = row[q];
    acc += v.x * v.x + v.y * v.y + v.z * v.z + v.w * v.w;
  }
  nrm[i] = sqrtf(acc);
}

// ---------------- kernel 2: KNN via V_WMMA_F32_16X16X4_F32 ----------------
// Block = 256 threads (8 waves) owns 16 query rows. Per iteration the 8 waves
// compute a 16x128 tile of scores  S[m][j] = sq_j - 2*pos_m.pos_j  (monotone in
// d2 per row) with one 16x16x4 f32 WMMA per wave, staged through a
// double-buffered LDS tile (one barrier per iteration); the 256 threads
// (16 rows x 16 column-lanes) fold the tile into private running top-3 state
// with branchless cndmask updates. One final 48-way merge per row at the end.
__global__ __launch_bounds__(256) void knn_kernel(const float* __restrict__ pos,
                                                  int* __restrict__ knn_idx) {
  __shared__ float S[2][16 * 132];  // double-buffered 16x128 tile, padded stride
  __shared__ float cs[16 * 48];     // per-row candidate scores (16 subs x 3)
  __shared__ int   ci[16 * 48];     // per-row candidate indices

  const int t    = threadIdx.x;
  const int wave = t >> 5;
  const int lane = t & 31;
  const int lh   = lane & 15;   // M (for A) / N (for B) within tile
  const int khi  = lane >> 4;   // which K-pair this lane holds
  const int rowBase = blockIdx.x << 4;

  // A fragment: row m = rowBase+lh, lane holds K = {2*khi, 2*khi+1}
  // A[m] = (-2x, -2y, -2z, 1)
  const int am = rowBase + lh;
  const float ax = pos[3 * am + 0];
  const float ay = pos[3 * am + 1];
  const float az = pos[3 * am + 2];
  v2f afrag;
  afrag.x = khi ? (-2.0f * az) : (-2.0f * ax);
  afrag.y = khi ? 1.0f         : (-2.0f * ay);

  // scan role: this thread scans row r, columns == sub (mod 16)
  const int r    = t >> 4;
  const int sub  = t & 15;
  const int grow = rowBase + r;
  float s0 = FBIG, s1 = FBIG, s2 = FBIG;
  int   i0 = -1, i1 = -1, i2 = -1;

  for (int ct = 0; ct < NPTS / 128; ++ct) {
    float* Sb = S[ct & 1];
    // B fragment: column j = ct*128 + wave*16 + lh; B[:,j] = (x, y, z, sq)
    const int bj = ct * 128 + wave * 16 + lh;
    const float bx = pos[3 * bj + 0];
    const float by = pos[3 * bj + 1];
    const float bz = pos[3 * bj + 2];
    const float sq = bx * bx + by * by + bz * bz;
    v2f bfrag;
    bfrag.x = khi ? bz : bx;
    bfrag.y = khi ? sq : by;

    v8f cacc = {0.f, 0.f, 0.f, 0.f, 0.f, 0.f, 0.f, 0.f};
    v8f d = __builtin_amdgcn_wmma_f32_16x16x4_f32(
        /*neg_a=*/false, afrag, /*neg_b=*/false, bfrag,
        /*c_mod=*/(short)0, cacc, /*reuse_a=*/false, /*reuse_b=*/false);

    // D layout: VGPR q holds M = q (lanes 0-15) or q+8 (lanes 16-31), N = lane%16
    const int rb = khi * 8;
    const int cc = wave * 16 + lh;
#pragma unroll
    for (int q = 0; q < 8; ++q) Sb[(rb + q) * 132 + cc] = d[q];

    // single barrier per iteration: orders this buffer's writes before its
    // scan, and (transitively) the previous scan of this buffer before the
    // next overwrite two iterations later.
    __syncthreads();

#pragma unroll
    for (int q = 0; q < 8; ++q) {
      const int cl = sub + (q << 4);
      const int gc = ct * 128 + cl;
      float s = Sb[r * 132 + cl];
      s = (gc == grow) ? FBIG : s;  // loop=False: exclude self, branchlessly
      top3_update(s, gc, s0, s1, s2, i0, i1, i2);
    }
  }
  __syncthreads();

  // final merge: 16 partial top-3 per row -> top-3
  cs[r * 48 + sub * 3 + 0] = s0; ci[r * 48 + sub * 3 + 0] = i0;
  cs[r * 48 + sub * 3 + 1] = s1; ci[r * 48 + sub * 3 + 1] = i1;
  cs[r * 48 + sub * 3 + 2] = s2; ci[r * 48 + sub * 3 + 2] = i2;
  __syncthreads();
  if (t < 16) {
    float m0 = FBIG, m1 = FBIG, m2 = FBIG;
    int   j0 = -1, j1 = -1, j2 = -1;
    for (int q = 0; q < 48; ++q) {
      const float s = cs[t * 48 + q];
      const int  id = ci[t * 48 + q];
      if (id >= 0) top3_update(s, id, m0, m1, m2, j0, j1, j2);
    }
    const int gr = rowBase + t;
    knn_idx[gr * 3 + 0] = j0;
    knn_idx[gr * 3 + 1] = j1;
    knn_idx[gr * 3 + 2] = j2;
  }
}

// ---------------- kernel 3: cosine filter -> keys + bucket counts ----------------
__global__ __launch_bounds__(256) void edge_kernel(const float* __restrict__ x,
                                                   const int* __restrict__ knn_idx,
                                                   const int* __restrict__ ei,
                                                   const float* __restrict__ nrm,
                                                   unsigned* __restrict__ keys,
                                                   unsigned* __restrict__ counts) {
  int t = blockIdx.x * blockDim.x + threadIdx.x;
  if (t >= M_TOT) return;
  int s, d;
  if (t < NPTS * KNN) { s = t / KNN; d = knn_idx[t]; }
  else { int e = t - NPTS * KNN; s = ei[e]; d = ei[NEDGE + e]; }

  const float4* xs = (const float4*)(x + (size_t)s * DIMF);
  const float4* xd = (const float4*)(x + (size_t)d * DIMF);
  float acc = 0.f;
#pragma unroll
  for (int q = 0; q < DIMF / 4; ++q) {
    float4 a = xs[q], b = xd[q];
    acc += a.x * b.x + a.y * b.y + a.z * b.z + a.w * b.w;
  }
  const float den = fmaxf(nrm[s] * nrm[d], COS_EPS);
  const bool valid = (acc / den) >= THRESH;

  keys[t]         = valid ? (((unsigned)s << 14) | (unsigned)d) : INVALID_KEY;
  keys[M_TOT + t] = valid ? (((unsigned)d << 14) | (unsigned)s) : INVALID_KEY;
  if (valid) {
    atomicAdd(&counts[s], 1u);
    atomicAdd(&counts[d], 1u);
  }
}

// ---------------- kernel 4: exclusive prefix sum over 16384 bucket counts ----------------
__global__ __launch_bounds__(256) void scan_kernel(const unsigned* __restrict__ counts,
                                                   unsigned* __restrict__ offsets) {
  __shared__ unsigned ps[256];
  const int t = threadIdx.x;
  unsigned sum = 0;
#pragma unroll 4
  for (int i = 0; i < 64; ++i) sum += counts[t * 64 + i];
  ps[t] = sum;
  __syncthreads();
  for (int off = 1; off < 256; off <<= 1) {
    unsigned v = (t >= off) ? ps[t - off] : 0u;
    __syncthreads();
    ps[t] += v;
    __syncthreads();
  }
  unsigned base = ps[t] - sum;  // exclusive base for this chunk
  for (int i = 0; i < 64; ++i) {
    offsets[t * 64 + i] = base;
    base += counts[t * 64 + i];
  }
}

// ---------------- kernel 5: scatter keys into src buckets ----------------
__global__ __launch_bounds__(256) void scatter_kernel(const unsigned* __restrict__ keys,
                                                      const unsigned* __restrict__ offsets,
                                                      unsigned* __restrict__ cursors,
                                                      unsigned* __restrict__ sorted) {
  int t = blockIdx.x * blockDim.x + threadIdx.x;
  if (t >= TWO_M) return;
  unsigned k = keys[t];
  if (k == INVALID_KEY) return;
  unsigned b = k >> 14;
  unsigned p = offsets[b] + atomicAdd(&cursors[b], 1u);
  sorted[p] = k;
}

// ---------------- kernel 6: per-bucket sort + positional dedup + emit ----------------
__global__ __launch_bounds__(256) void bucket_kernel(const unsigned* __restrict__ counts,
                                                     const unsigned* __restrict__ offsets,
                                                     unsigned* __restrict__ sorted,
                                                     int* __restrict__ out) {
  int b = blockIdx.x * blockDim.x + threadIdx.x;
  if (b >= NPTS) return;
  const unsigned off = offsets[b];
  const unsigned cnt = counts[b];
  // insertion sort ascending by dst (all keys here share src = b); L2-resident
  for (unsigned i = 1; i < cnt; ++i) {
    unsigned k = sorted[off + i];
    int j = (int)i - 1;
    while (j >= 0 && sorted[off + (unsigned)j] > k) {
      sorted[off + (unsigned)j + 1] = sorted[off + (unsigned)j];
      --j;
    }
    sorted[off + (unsigned)j + 1] = k;
  }
  // dedup is bucket-local (bucket boundary => different src => different key);
  // duplicate slots stay -1 (out pre-initialized), matching jnp.where semantics.
  unsigned prev = INVALID_KEY;
  for (unsigned i = 0; i < cnt; ++i) {
    unsigned k = sorted[off + i];
    if (k != prev) {
      out[off + i]         = (int)(k >> 14);
      out[TWO_M + off + i] = (int)(k & 16383u);
    }
    prev = k;
  }
}

// ---------------- launch ----------------
extern "C" void kernel_launch(void* const* d_in, const int* in_sizes, int n_in,
                              void* d_out, int out_size, void* d_ws, size_t ws_size,
                              hipStream_t stream) {
  const float* x   = (const float*)d_in[0];   // [N, 64]
  const float* pos = (const float*)d_in[1];   // [N, 3]
  const int*   ei  = (const int*)d_in[2];     // [2, E]
  int* out = (int*)d_out;                     // [2, 2M] int32

  // workspace layout (bytes); total ~5.2 MB
  char* ws = (char*)d_ws;
  float*    nrm     = (float*)(ws + 0);                 // 64 KB
  int*      knn_idx = (int*)(ws + 65536);               // 192 KB
  unsigned* keys    = (unsigned*)(ws + 262144);         // 2.4 MB
  unsigned* counts  = (unsigned*)(ws + 2752512);        // 64 KB
  unsigned* offsets = (unsigned*)(ws + 2818048);        // 64 KB
  unsigned* cursors = (unsigned*)(ws + 2883584);        // 64 KB
  unsigned* sorted  = (unsigned*)(ws + 2949120);        // 2.4 MB

  init_kernel<<<(OUT_ELEMS + 255) / 256, 256, 0, stream>>>(out, counts, cursors);
  norm_kernel<<<(NPTS + 255) / 256, 256, 0, stream>>>(x, nrm);
  knn_kernel<<<NPTS / 16, 256, 0, stream>>>(pos, knn_idx);
  edge_kernel<<<(M_TOT + 255) / 256, 256, 0, stream>>>(x, knn_idx, ei, nrm, keys, counts);
  scan_kernel<<<1, 256, 0, stream>>>(counts, offsets);
  scatter_kernel<<<(TWO_M + 255) / 256, 256, 0, stream>>>(keys, offsets, cursors, sorted);
  bucket_kernel<<<(NPTS + 255) / 256, 256, 0, stream>>>(counts, offsets, sorted, out);
}